// HomoGNN_GraphConv_39977555591469
// MI455X (gfx1250) — compile-verified
//
#include <hip/hip_runtime.h>
#include <hip/hip_bf16.h>

// CDNA5 / gfx1250 fused GraphConv x2 + mean-pool + linear.
// fp32 WMMA (v_wmma_f32_16x16x4_f32): keeps reference fp32 numerics; the
// workload is scatter/atomic bandwidth bound so there is no reason to drop
// to bf16/fp8 for the tiny (5 GFLOP) GEMM portion.

#define DM 128            // feature dim
#define NTILES 8          // 128 / 16 output column tiles per wave

typedef float v2f __attribute__((ext_vector_type(2)));
typedef float v8f __attribute__((ext_vector_type(8)));

__device__ __forceinline__ v8f wmma4(v2f a, v2f b, v8f c) {
    // 8 args: (neg_a, A, neg_b, B, c_mod, C, reuse_a, reuse_b)
    return __builtin_amdgcn_wmma_f32_16x16x4_f32(
        false, a, false, b, (short)0, c, false, false);
}

// ---------------------------------------------------------------- utilities
__global__ void zero_f32(float* __restrict__ p, int n) {
    int i = blockIdx.x * blockDim.x + threadIdx.x;
    if (i < n) p[i] = 0.0f;
}

__global__ void edge_degree(const int* __restrict__ dst, float* __restrict__ deg, int nE) {
    int i = blockIdx.x * blockDim.x + threadIdx.x;
    if (i < nE) atomicAdd(&deg[dst[i]], 1.0f);
}

__global__ void batch_count(const int* __restrict__ batch, float* __restrict__ cnt, int nN) {
    int i = blockIdx.x * blockDim.x + threadIdx.x;
    if (i < nN) atomicAdd(&cnt[batch[i]], 1.0f);
}

// Row scatter-add: 32 threads per edge, float4 per thread (one 512B row / wave).
__global__ void scatter_rows(const int* __restrict__ src, const int* __restrict__ dst,
                             const float* __restrict__ in, float* __restrict__ agg, int nE) {
    int idx = blockIdx.x * blockDim.x + threadIdx.x;
    if (idx >= nE * 32) return;
    int e = idx >> 5;
    int c = (idx & 31) << 2;     // float offset within the 128-float row
    int s = src[e], d = dst[e];
    const float4 v = *(const float4*)(in + (long long)s * DM + c);
    float* p = agg + (long long)d * DM + c;
    atomicAdd(p + 0, v.x);
    atomicAdd(p + 1, v.y);
    atomicAdd(p + 2, v.z);
    atomicAdd(p + 3, v.w);
}

// ------------------------------------------------- WMMA tile core (16x128)
// Accumulate acc[8] (16x16 f32 tiles) with A-rows scaled on the fly.
// A fragment: lane (0..15 | 16..31) -> K halves (k,k+1 | k+2,k+3), M = lane&15.
// B fragment: B[k][n] = W[n][k]; same K-half split across lane halves.
__device__ __forceinline__ void gemm_part(const float* __restrict__ A, float ascale,
                                          const float* __restrict__ W,
                                          int arow, int half, int l16, v8f acc[NTILES]) {
    for (int k = 0; k < DM; k += 4) {
        const int kk = k + 2 * half;
        float2 av = *(const float2*)(A + (long long)arow * DM + kk);
        v2f a;
        a.x = av.x * ascale;
        a.y = av.y * ascale;
#pragma unroll
        for (int t = 0; t < NTILES; ++t) {
            const int ncol = t * 16 + l16;
            float2 bv = *(const float2*)(W + ncol * DM + kk);
            v2f b;
            b.x = bv.x;
            b.y = bv.y;
            acc[t] = wmma4(a, b, acc[t]);
        }
    }
}

// h = relu(agg/deg @ Wrel^T + x @ Wroot^T + b)
__global__ __launch_bounds__(256)
void layer_wmma(const float* __restrict__ agg, const float* __restrict__ deg,
                const float* __restrict__ xin,
                const float* __restrict__ Wrel, const float* __restrict__ Wroot,
                const float* __restrict__ bias, float* __restrict__ out, int nRows) {
    const int lane = threadIdx.x & 31;
    const int wave = threadIdx.x >> 5;
    const int tile = blockIdx.x * 8 + wave;       // 16-row tile (wave-uniform)
    const int m0 = tile * 16;
    if (m0 >= nRows) return;                      // whole wave exits together
    const int half = lane >> 4;
    const int l16 = lane & 15;
    const int arow = m0 + l16;
    const float dscale = 1.0f / fmaxf(deg[arow], 1.0f);

    v8f acc[NTILES];
#pragma unroll
    for (int t = 0; t < NTILES; ++t) acc[t] = (v8f){};

    gemm_part(agg, dscale, Wrel, arow, half, l16, acc);   // mean-aggregated term
    gemm_part(xin, 1.0f, Wroot, arow, half, l16, acc);    // root term

#pragma unroll
    for (int t = 0; t < NTILES; ++t) {
        const int ncol = t * 16 + l16;
        const float bv = bias[ncol];
#pragma unroll
        for (int r = 0; r < 8; ++r) {
            const int row = m0 + r + 8 * half;
            out[(long long)row * DM + ncol] = fmaxf(acc[t][r] + bv, 0.0f);
        }
    }
}

// Layer 2 fused with pooling: relu row results atomically accumulated into
// per-graph sums (global_mean_pool numerator), skipping the h2 buffer.
__global__ __launch_bounds__(256)
void layer2_pool_wmma(const float* __restrict__ agg, const float* __restrict__ deg,
                      const float* __restrict__ hin,
                      const float* __restrict__ Wrel, const float* __restrict__ Wroot,
                      const float* __restrict__ bias, const int* __restrict__ batch,
                      float* __restrict__ sums, int nRows) {
    const int lane = threadIdx.x & 31;
    const int wave = threadIdx.x >> 5;
    const int tile = blockIdx.x * 8 + wave;
    const int m0 = tile * 16;
    if (m0 >= nRows) return;
    const int half = lane >> 4;
    const int l16 = lane & 15;
    const int arow = m0 + l16;
    const float dscale = 1.0f / fmaxf(deg[arow], 1.0f);

    v8f acc[NTILES];
#pragma unroll
    for (int t = 0; t < NTILES; ++t) acc[t] = (v8f){};

    gemm_part(agg, dscale, Wrel, arow, half, l16, acc);
    gemm_part(hin, 1.0f, Wroot, arow, half, l16, acc);

    int grp[8];
#pragma unroll
    for (int r = 0; r < 8; ++r) grp[r] = batch[m0 + r + 8 * half];

#pragma unroll
    for (int t = 0; t < NTILES; ++t) {
        const int ncol = t * 16 + l16;
        const float bv = bias[ncol];
#pragma unroll
        for (int r = 0; r < 8; ++r) {
            const float v = fmaxf(acc[t][r] + bv, 0.0f);
            atomicAdd(&sums[grp[r] * DM + ncol], v);
        }
    }
}

// out = (sums/cnt) @ Wc^T + bc    (G=64 rows -> 4 wave tiles, one block)
__global__ __launch_bounds__(128)
void pool_gemm(const float* __restrict__ sums, const float* __restrict__ cnt,
               const float* __restrict__ Wc, const float* __restrict__ bc,
               float* __restrict__ out, int nG) {
    const int lane = threadIdx.x & 31;
    const int wave = threadIdx.x >> 5;
    const int m0 = wave * 16;
    if (m0 >= nG) return;
    const int half = lane >> 4;
    const int l16 = lane & 15;
    const int arow = m0 + l16;
    const float dscale = 1.0f / fmaxf(cnt[arow], 1.0f);

    v8f acc[NTILES];
#pragma unroll
    for (int t = 0; t < NTILES; ++t) acc[t] = (v8f){};

    gemm_part(sums, dscale, Wc, arow, half, l16, acc);

#pragma unroll
    for (int t = 0; t < NTILES; ++t) {
        const int ncol = t * 16 + l16;
        const float bv = bc[ncol];
#pragma unroll
        for (int r = 0; r < 8; ++r) {
            const int row = m0 + r + 8 * half;
            out[row * DM + ncol] = acc[t][r] + bv;
        }
    }
}

// ---------------------------------------------------------------- launcher
extern "C" void kernel_launch(void* const* d_in, const int* in_sizes, int n_in,
                              void* d_out, int out_size, void* d_ws, size_t ws_size,
                              hipStream_t stream) {
    const float* x      = (const float*)d_in[0];
    const int*   ei     = (const int*)d_in[1];    // [2, E] flat: src then dst
    const int*   batch  = (const int*)d_in[2];
    const float* Wrel1  = (const float*)d_in[3];
    const float* Wroot1 = (const float*)d_in[4];
    const float* b1     = (const float*)d_in[5];
    const float* Wrel2  = (const float*)d_in[6];
    const float* Wroot2 = (const float*)d_in[7];
    const float* b2     = (const float*)d_in[8];
    const float* Wc     = (const float*)d_in[9];
    const float* bc     = (const float*)d_in[10];

    const int N = in_sizes[0] / DM;       // 40000
    const int E = in_sizes[1] / 2;        // 640000
    const int G = out_size / DM;          // 64
    const int* src = ei;
    const int* dst = ei + E;

    // workspace layout
    float* agg  = (float*)d_ws;           // N*DM
    float* deg  = agg + (long long)N * DM;// N
    float* h    = deg + N;                // N*DM
    float* sums = h + (long long)N * DM;  // G*DM
    float* cnt  = sums + G * DM;          // G
    const int wsElems = N * DM * 2 + N + G * DM + G;

    const int TPB = 256;
    // 1) zero all scratch
    zero_f32<<<(wsElems + TPB - 1) / TPB, TPB, 0, stream>>>(agg, wsElems);
    // 2) degrees + per-graph node counts
    edge_degree<<<(E + TPB - 1) / TPB, TPB, 0, stream>>>(dst, deg, E);
    batch_count<<<(N + TPB - 1) / TPB, TPB, 0, stream>>>(batch, cnt, N);
    // 3) layer 1: scatter + fused GEMM/bias/relu
    scatter_rows<<<(E * 32 + TPB - 1) / TPB, TPB, 0, stream>>>(src, dst, x, agg, E);
    const int rowTiles = (N + 15) / 16;
    const int gemmBlocks = (rowTiles + 7) / 8;
    layer_wmma<<<gemmBlocks, TPB, 0, stream>>>(agg, deg, x, Wrel1, Wroot1, b1, h, N);
    // 4) layer 2: re-zero agg, scatter h, fused GEMM + pooling accumulation
    zero_f32<<<(N * DM + TPB - 1) / TPB, TPB, 0, stream>>>(agg, N * DM);
    scatter_rows<<<(E * 32 + TPB - 1) / TPB, TPB, 0, stream>>>(src, dst, h, agg, E);
    layer2_pool_wmma<<<gemmBlocks, TPB, 0, stream>>>(agg, deg, h, Wrel2, Wroot2, b2,
                                                     batch, sums, N);
    // 5) final linear on pooled means
    pool_gemm<<<1, 128, 0, stream>>>(sums, cnt, Wc, bc, (float*)d_out, G);
}